// down_17867063951705
// MI455X (gfx1250) — compile-verified
//
#include <hip/hip_runtime.h>
#include <hip/hip_bf16.h>

typedef __attribute__((ext_vector_type(2))) float v2f;
typedef __attribute__((ext_vector_type(8))) float v8f;
typedef __attribute__((ext_vector_type(4))) unsigned int u32x4;
typedef __attribute__((ext_vector_type(8))) int i32x8;
typedef __attribute__((ext_vector_type(4))) int i32x4;

#define IN_C   32
#define OUT_C  64
#define NK     27          // K**DIM spline kernels
#define NK1    28          // + root slot
#define MAXNEI 64

// ---------------------------------------------------------------------------
// Zero-init agg + deg (contiguous region)
// ---------------------------------------------------------------------------
__global__ void init_zero_kernel(float* __restrict__ p, int n) {
    int t = blockIdx.x * blockDim.x + threadIdx.x;
    if (t < n) p[t] = 0.0f;
}

// ---------------------------------------------------------------------------
// T[n,k,o] = x[n,:] @ W[k,:,:]   (k==27 -> root)   via V_WMMA_F32_16X16X4_F32.
// One block handles one spline kernel k for 32 nodes: 8 waves = 2 node-tiles
// x 4 out-tiles. W[k] (32x64 f32 = 8KB) is staged into LDS once per block by
// the Tensor Data Mover (TENSOR_LOAD_TO_LDS, TENSORcnt), then all 8 waves
// read their B fragments from LDS.
//
// Fragment layouts (ISA 7.12.2, f32):
//   A 16x4 (2 VGPRs): lane<16 -> row=lane, K=0/1; lane>=16 -> K=2/3
//   B 4x16 (2 VGPRs): lane<16 -> col=lane, K=0/1; lane>=16 -> K=2/3
//   D 16x16 (8 VGPRs): vgpr j -> row j (+8 for lanes>=16), col = lane&15
// ---------------------------------------------------------------------------
__global__ void __launch_bounds__(256) spline_T_wmma_kernel(
    const float* __restrict__ x,     // [N, 32]
    const float* __restrict__ W,     // [27, 32, 64]
    const float* __restrict__ root,  // [32, 64]
    float* __restrict__ T,           // [N, 28, 64]
    int N)
{
    __shared__ float sW[IN_C * OUT_C];       // 8 KB; sole LDS object -> offset 0

    const int kidx  = blockIdx.x % NK1;      // spline kernel (27 == root)
    const int mpair = blockIdx.x / NK1;      // pair of 16-node tiles
    const float* Wk = (kidx < NK) ? (W + (size_t)kidx * IN_C * OUT_C) : root;

#if __has_builtin(__builtin_amdgcn_tensor_load_to_lds)
    // --- TDM: DMA the 32x64 f32 tile of W[k] into LDS (D# per ISA ch.8) ---
    // 6-arg toolchain form: (g0 u32x4, g1 i32x8, g2 i32x4, g3 i32x4, i32x8, cpol)
    if (threadIdx.x < 32) {                  // wave 0 issues; EXEC ignored by TDM
        unsigned long long ga = (unsigned long long)(uintptr_t)Wk;
        u32x4 g0;
        g0.x = 1u;                                        // count=1 (valid user D#)
        g0.y = 0u;                                        // lds_addr = 0 (sW base)
        g0.z = (unsigned)(ga & 0xffffffffu);              // global_addr[31:0]
        g0.w = (unsigned)((ga >> 32) & 0x01ffffffu)       // global_addr[56:32]
             | (2u << 30);                                // type = 2 ("image")
        i32x8 g1;
        g1[0] = 0x00020000;                               // data_size=4B; mask=0
        g1[1] = (OUT_C & 0xffff) << 16;                   // tensor_dim0 lo16
        g1[2] = (OUT_C >> 16) | ((IN_C & 0xffff) << 16);  // dim0 hi | dim1 lo
        g1[3] = (IN_C >> 16) | (OUT_C << 16);             // dim1 hi | tile_dim0
        g1[4] = IN_C;                                     // tile_dim1 (tile_dim2=0)
        g1[5] = OUT_C;                                    // tensor_dim0_stride lo
        g1[6] = 0;                                        // stride0 hi | stride1 lo
        g1[7] = 0;                                        // stride1 hi
        i32x4 g2 = {0, 0, 0, 0};                          // 2D tensor: groups 2/3
        i32x4 g3 = {0, 0, 0, 0};                          //   unused (zeros)
        i32x8 g4 = {0, 0, 0, 0, 0, 0, 0, 0};              // extra group (zeros)
        __builtin_amdgcn_tensor_load_to_lds(g0, g1, g2, g3, g4, 0);
        __builtin_amdgcn_s_wait_tensorcnt(0);             // s_wait_tensorcnt 0x0
    }
    __syncthreads();
    // The TDM write happens behind the compiler's back. Escape sW's address
    // into opaque asm so the compiler must assume it was written -> keeps the
    // LDS object alive and forces real ds_loads of the B fragments below.
    asm volatile("" :: "v"(sW) : "memory");
#else
    for (int i = threadIdx.x; i < IN_C * OUT_C; i += 256) sW[i] = Wk[i];
    __syncthreads();
#endif

    const int lane  = threadIdx.x & 31;
    const int wave  = threadIdx.x >> 5;
    const int otile = wave & 3;              // 16-wide slice of OUT_C
    const int mtile = mpair * 2 + (wave >> 2);
    if (mtile >= N / 16) return;

    const int rc   = lane & 15;              // A row / B col
    const int kOff = (lane >> 4) * 2;        // 0 or 2

    const float* xrow = x + (size_t)(mtile * 16 + rc) * IN_C;
    const float* wcol = sW + otile * 16 + rc;

    v8f acc = {};
    #pragma unroll
    for (int kb = 0; kb < IN_C; kb += 4) {
        v2f a, b;
        a.x = xrow[kb + kOff];
        a.y = xrow[kb + kOff + 1];
        b.x = wcol[(kb + kOff) * OUT_C];     // ds_load from staged tile
        b.y = wcol[(kb + kOff + 1) * OUT_C];
        acc = __builtin_amdgcn_wmma_f32_16x16x4_f32(
            /*neg_a=*/false, a, /*neg_b=*/false, b,
            /*c_mod=*/(short)0, acc, /*reuse_a=*/false, /*reuse_b=*/false);
    }

    const int ncol = lane & 15;
    const int mAdd = (lane >> 4) * 8;
    #pragma unroll
    for (int j = 0; j < 8; ++j) {
        int m = mtile * 16 + j + mAdd;
        T[((size_t)m * NK1 + kidx) * OUT_C + otile * 16 + ncol] = acc[j];
    }
}

// ---------------------------------------------------------------------------
// Per-edge spline gather + mean-aggregation scatter.
// 4 threads per edge, 16 out-channels each. T rows are L2-resident (117MB<192MB).
// ---------------------------------------------------------------------------
__global__ void __launch_bounds__(256) edge_gather_kernel(
    const int*   __restrict__ ei,    // [2, E]
    const float* __restrict__ ea,    // [E, 3]
    const float* __restrict__ T,     // [N, 28, 64]
    float* __restrict__ agg,         // [N, 64]
    float* __restrict__ deg,         // [N]
    int E)
{
    int t = blockIdx.x * blockDim.x + threadIdx.x;
    int e = t >> 2;
    int chunk = t & 3;
    if (e >= E) return;

    int src = ei[e];
    int dst = ei[E + e];

    float f[3]; int i0[3];
    #pragma unroll
    for (int d = 0; d < 3; ++d) {
        float v = ea[(size_t)e * 3 + d] * 2.0f;   // (K-1) = 2
        float fl = floorf(v);
        f[d]  = v - fl;
        i0[d] = (int)fl;
    }

    const float* Tsrc = T + (size_t)src * NK1 * OUT_C;
    __builtin_prefetch(Tsrc, 0, 1);               // global_prefetch_b8

    float acc[16];
    #pragma unroll
    for (int j = 0; j < 16; ++j) acc[j] = 0.0f;

    #pragma unroll
    for (int b = 0; b < 8; ++b) {
        float basis = 1.0f;
        int widx = 0, pw = 1;
        #pragma unroll
        for (int d = 0; d < 3; ++d) {
            int bit = (b >> d) & 1;
            basis *= bit ? f[d] : (1.0f - f[d]);
            widx  += (i0[d] + bit) * pw;
            pw *= 3;
        }
        const float* Trow = Tsrc + (size_t)widx * OUT_C + chunk * 16;
        #pragma unroll
        for (int j = 0; j < 16; ++j) acc[j] += basis * Trow[j];
    }

    float* ap = agg + (size_t)dst * OUT_C + chunk * 16;
    #pragma unroll
    for (int j = 0; j < 16; ++j) atomicAdd(ap + j, acc[j]);
    if (chunk == 0) atomicAdd(deg + dst, 1.0f);
}

// ---------------------------------------------------------------------------
// h = ELU(agg/max(deg,1) + x@root + bias); x@root lives in T slot k=27
// ---------------------------------------------------------------------------
__global__ void __launch_bounds__(256) finalize_kernel(
    const float* __restrict__ T,
    const float* __restrict__ agg,
    const float* __restrict__ deg,
    const float* __restrict__ bias,
    float* __restrict__ hbuf,        // [N, 64]
    int N)
{
    int t = blockIdx.x * blockDim.x + threadIdx.x;
    if (t >= N * OUT_C) return;
    int n = t >> 6, o = t & 63;
    float dv = deg[n];
    dv = dv > 1.0f ? dv : 1.0f;
    float hv = agg[t] / dv + T[((size_t)n * NK1 + NK) * OUT_C + o] + bias[o];
    hbuf[t] = hv > 0.0f ? hv : (__expf(hv) - 1.0f);
}

// ---------------------------------------------------------------------------
// Farthest point sampling: one workgroup, running min-dist array in LDS.
// ---------------------------------------------------------------------------
__global__ void __launch_bounds__(1024) fps_kernel(
    const float* __restrict__ pos, int* __restrict__ idx, int N, int S)
{
    extern __shared__ float smem[];
    float* d  = smem;                      // [N]
    float* rv = smem + N;                  // [1024]
    int*   ri = (int*)(rv + 1024);         // [1024]
    const int tid = threadIdx.x;

    float px0 = pos[0], py0 = pos[1], pz0 = pos[2];
    for (int i = tid; i < N; i += 1024) {
        float dx = pos[(size_t)i * 3 + 0] - px0;
        float dy = pos[(size_t)i * 3 + 1] - py0;
        float dz = pos[(size_t)i * 3 + 2] - pz0;
        d[i] = dx * dx + dy * dy + dz * dz;
    }
    if (tid == 0) idx[0] = 0;
    __syncthreads();

    for (int s = 1; s < S; ++s) {
        float best = -1.0f; int bi = 0;
        for (int i = tid; i < N; i += 1024) {
            float v = d[i];
            if (v > best) { best = v; bi = i; }
        }
        rv[tid] = best; ri[tid] = bi;
        __syncthreads();
        for (int off = 512; off > 0; off >>= 1) {
            if (tid < off && rv[tid + off] > rv[tid]) {
                rv[tid] = rv[tid + off];
                ri[tid] = ri[tid + off];
            }
            __syncthreads();
        }
        int sel = ri[0];
        if (tid == 0) idx[s] = sel;
        float qx = pos[(size_t)sel * 3 + 0];
        float qy = pos[(size_t)sel * 3 + 1];
        float qz = pos[(size_t)sel * 3 + 2];
        for (int i = tid; i < N; i += 1024) {
            float dx = pos[(size_t)i * 3 + 0] - qx;
            float dy = pos[(size_t)i * 3 + 1] - qy;
            float dz = pos[(size_t)i * 3 + 2] - qz;
            float v = dx * dx + dy * dy + dz * dz;
            if (v < d[i]) d[i] = v;
        }
        __syncthreads();
    }
}

// ---------------------------------------------------------------------------
// radius(): per query, 64 nearest (iterative keyed-min over LDS d2 array),
// masked by r^2. Writes edge_index straight into d_out.
// ---------------------------------------------------------------------------
__global__ void __launch_bounds__(256) radius_kernel(
    const float* __restrict__ pos,
    const int*   __restrict__ idx,
    float* __restrict__ out,         // d_out base
    int N, int S, float r2)
{
    extern __shared__ float sd[];    // [N] d2
    unsigned long long* rk = (unsigned long long*)(sd + N);   // [256]
    const int q = blockIdx.x;
    const int tid = threadIdx.x;

    int qi = idx[q];
    float qx = pos[(size_t)qi * 3 + 0];
    float qy = pos[(size_t)qi * 3 + 1];
    float qz = pos[(size_t)qi * 3 + 2];
    for (int i = tid; i < N; i += 256) {
        float dx = pos[(size_t)i * 3 + 0] - qx;
        float dy = pos[(size_t)i * 3 + 1] - qy;
        float dz = pos[(size_t)i * 3 + 2] - qz;
        sd[i] = dx * dx + dy * dy + dz * dz;
    }
    __syncthreads();

    int* outCol = (int*)out + (size_t)S * OUT_C;               // after h
    int* outRow = outCol + (size_t)S * MAXNEI;

    for (int j = 0; j < MAXNEI; ++j) {
        unsigned long long best = ~0ull;
        for (int i = tid; i < N; i += 256) {
            unsigned u = __float_as_uint(sd[i]);               // non-neg: order-preserving
            unsigned long long key = ((unsigned long long)u << 32) | (unsigned)i;
            if (key < best) best = key;
        }
        rk[tid] = best;
        __syncthreads();
        for (int off = 128; off > 0; off >>= 1) {
            if (tid < off && rk[tid + off] < rk[tid]) rk[tid] = rk[tid + off];
            __syncthreads();
        }
        unsigned long long sel = rk[0];
        __syncthreads();
        int si = (int)(sel & 0xffffffffu);
        float sv = __uint_as_float((unsigned)(sel >> 32));
        if (tid == 0) {
            bool ok = sv <= r2;
            outCol[(size_t)q * MAXNEI + j] = ok ? si : -1;
            outRow[(size_t)q * MAXNEI + j] = ok ? q  : -1;
        }
        if (tid == (si & 255)) sd[si] = __builtin_huge_valf(); // knock out selected
        __syncthreads();
    }
}

// ---------------------------------------------------------------------------
// Output gathers: h[idx], pos[idx], batch[idx], edge_attr[idx]
// ---------------------------------------------------------------------------
__global__ void __launch_bounds__(256) gather_kernel(
    const float* __restrict__ hbuf,
    const float* __restrict__ pos,
    const int*   __restrict__ batch,
    const float* __restrict__ ea,
    const int*   __restrict__ idx,
    float* __restrict__ out, int S)
{
    int t = blockIdx.x * blockDim.x + threadIdx.x;
    const size_t posBase   = (size_t)S * OUT_C + 2ull * S * MAXNEI;
    const size_t batchBase = posBase + (size_t)S * 3;
    const size_t eaBase    = batchBase + (size_t)S;

    if (t < S * OUT_C) {
        int q = t >> 6, o = t & 63;
        __builtin_prefetch(hbuf + (size_t)idx[q] * OUT_C, 0, 0);
        out[t] = hbuf[(size_t)idx[q] * OUT_C + o];
    }
    if (t < S * 3) {
        int q = t / 3, d = t % 3;
        out[posBase + t] = pos[(size_t)idx[q] * 3 + d];
        out[eaBase + t]  = ea[(size_t)idx[q] * 3 + d];
    }
    if (t < S) {
        ((int*)out)[batchBase + t] = batch[idx[t]];
    }
}

// ---------------------------------------------------------------------------
extern "C" void kernel_launch(void* const* d_in, const int* in_sizes, int n_in,
                              void* d_out, int out_size, void* d_ws, size_t ws_size,
                              hipStream_t stream) {
    const float* x     = (const float*)d_in[0];
    const float* pos   = (const float*)d_in[1];
    const int*   ei    = (const int*)  d_in[2];
    const float* ea    = (const float*)d_in[3];
    const int*   batch = (const int*)  d_in[4];
    const float* W     = (const float*)d_in[5];
    const float* root  = (const float*)d_in[6];
    const float* bias  = (const float*)d_in[7];

    const int N = in_sizes[0] / IN_C;          // 16384
    const int E = in_sizes[3] / 3;             // 262144
    const int S = N / 4;                       // 4096 (RATIO=0.25)

    float* T    = (float*)d_ws;                          // [N,28,64]
    float* agg  = T   + (size_t)N * NK1 * OUT_C;         // [N,64]
    float* deg  = agg + (size_t)N * OUT_C;               // [N]
    float* hbuf = deg + N;                               // [N,64]
    int*   idx  = (int*)(hbuf + (size_t)N * OUT_C);      // [S]
    float* out  = (float*)d_out;

    // 1) zero agg + deg (contiguous)
    int zn = N * OUT_C + N;
    init_zero_kernel<<<(zn + 255) / 256, 256, 0, stream>>>(agg, zn);

    // 2) T = x @ W[k] for all 27 kernels + root slot (TDM-staged B + WMMA f32)
    int blocks = NK1 * (N / 32);
    spline_T_wmma_kernel<<<blocks, 256, 0, stream>>>(x, W, root, T, N);

    // 3) per-edge spline gather + scatter
    edge_gather_kernel<<<(E * 4 + 255) / 256, 256, 0, stream>>>(ei, ea, T, agg, deg, E);

    // 4) mean + root + bias + ELU
    finalize_kernel<<<(N * OUT_C + 255) / 256, 256, 0, stream>>>(T, agg, deg, bias, hbuf, N);

    // 5) FPS (single workgroup, LDS-resident distance array)
    size_t fps_smem = (size_t)(N + 1024) * sizeof(float) + 1024 * sizeof(int);
    fps_kernel<<<1, 1024, fps_smem, stream>>>(pos, idx, N, S);

    // 6) radius graph -> writes edge_index part of d_out directly
    size_t rad_smem = (size_t)N * sizeof(float) + 256 * sizeof(unsigned long long);
    radius_kernel<<<S, 256, rad_smem, stream>>>(pos, idx, out, N, S, 0.1f * 0.1f);

    // 7) output gathers
    gather_kernel<<<(S * OUT_C + 255) / 256, 256, 0, stream>>>(hbuf, pos, batch, ea, idx, out, S);
}